// GraphConvBlock_61314953117796
// MI455X (gfx1250) — compile-verified
//
#include <hip/hip_runtime.h>
#include <cstdint>

typedef __attribute__((ext_vector_type(2))) float v2f;
typedef __attribute__((ext_vector_type(8))) float v8f;

#define D 64
#define BN_EPS 1e-5f
#define WROW 160   // LDS pair-row stride in dwords (128 data + 32 pad -> bank-conflict-free)

// ---------------------------------------------------------------------------
// Workspace layout (floats):
//   dinv  : [0, 102400)                (N=100000 rounded up; deg then rsqrt(deg))
//   xw    : [102400, 102400+N*64)      x @ W
//   h     : [+N*64]                    aggregated features
//   sums  : [128]                      BN sum / sumsq per channel
// ---------------------------------------------------------------------------

__global__ void k_init(float* __restrict__ deg, float* __restrict__ sums, int N) {
    int gid = blockIdx.x * blockDim.x + threadIdx.x;
    if (gid < N) deg[gid] = 1.0f;          // self-loop contributes 1 to degree
    if (gid < 128) sums[gid] = 0.0f;       // BN accumulators
}

__global__ void k_count_deg(const long long* __restrict__ dst, float* __restrict__ deg, int E) {
    int gid = blockIdx.x * blockDim.x + threadIdx.x;
    if (gid >= E) return;
    atomicAdd(&deg[(int)dst[gid]], 1.0f);
}

__global__ void k_rsqrt(float* __restrict__ dinv, int N) {
    int gid = blockIdx.x * blockDim.x + threadIdx.x;
    if (gid < N) dinv[gid] = rsqrtf(dinv[gid]);
}

// ---------------------------------------------------------------------------
// xw = x @ W via V_WMMA_F32_16X16X4_F32.
// Wave computes a 16x64 slab (4 tiles of 16x16), K=64 in 16 steps of 4.
// A layout (16x4 f32): lanes 0-15 = M rows; VGPR0 = K{0|2}, VGPR1 = K{1|3}
// split by half-wave. B symmetric (N in place of M).
// W is staged K-pair-interleaved in LDS:
//   Wl[(k>>1)*WROW + col*2 + (k&1)] = W[k][col]
// so each B fragment is ONE aligned ds_load_b64, and the WROW=160 pad puts
// the two half-waves' pair-rows in disjoint bank halves.
// ---------------------------------------------------------------------------
__global__ void k_gemm_wmma(const float* __restrict__ x, const float* __restrict__ W,
                            float* __restrict__ xw, int N) {
    __shared__ float Wl[32 * WROW];
    for (int i = threadIdx.x; i < D * D; i += blockDim.x) {
        int k = i >> 6, col = i & 63;
        Wl[(k >> 1) * WROW + col * 2 + (k & 1)] = W[i];
    }
    __syncthreads();

    const int wave = threadIdx.x >> 5;
    const int lane = threadIdx.x & 31;
    const int rowBase = (blockIdx.x * 8 + wave) * 16;     // wave-uniform
    if (rowBase >= N) return;                             // uniform exit: EXEC stays all-1s

    const int half = lane >> 4;      // 0: lanes 0-15, 1: lanes 16-31
    const int lm   = lane & 15;

    int aRow = rowBase + lm;
    if (aRow >= N) aRow = N - 1;                          // clamp (stores masked later)
    const float* __restrict__ arow = x + (size_t)aRow * D;

    v8f acc[4];
#pragma unroll
    for (int t = 0; t < 4; ++t) acc[t] = 0;

#pragma unroll
    for (int k = 0; k < D; k += 4) {
        const int ka = k + half * 2;                      // even
        v2f a = *(const v2f*)(arow + ka);                 // A: (K=ka, K=ka+1), row aRow
        const float* brow = Wl + (ka >> 1) * WROW + lm * 2;
#pragma unroll
        for (int t = 0; t < 4; ++t) {
            v2f b = *(const v2f*)(brow + t * 32);         // (W[ka][col], W[ka+1][col])
            acc[t] = __builtin_amdgcn_wmma_f32_16x16x4_f32(
                false, a, false, b, (short)0, acc[t], false, false);
        }
    }

#pragma unroll
    for (int t = 0; t < 4; ++t) {
#pragma unroll
        for (int v = 0; v < 8; ++v) {
            int R = rowBase + half * 8 + v;
            if (R < N) xw[(size_t)R * D + t * 16 + lm] = acc[t][v];
        }
    }
}

// h[i][c] = b[c] + dinv[i]^2 * xw[i][c]   (bias + self-loop term)
__global__ void k_init_h(const float* __restrict__ xw, const float* __restrict__ dinv,
                         const float* __restrict__ bias, float* __restrict__ h, int N) {
    int gid = blockIdx.x * blockDim.x + threadIdx.x;
    if (gid >= N * 16) return;
    int i = gid >> 4;
    int c = (gid & 15) * 4;
    float s = dinv[i]; s = s * s;
    float4 xv = *(const float4*)(xw + (size_t)i * D + c);
    float4 bv = *(const float4*)(bias + c);
    float4 o;
    o.x = bv.x + s * xv.x; o.y = bv.y + s * xv.y;
    o.z = bv.z + s * xv.z; o.w = bv.w + s * xv.w;
    *(float4*)(h + (size_t)i * D + c) = o;
}

// Scatter-add: 16 consecutive lanes cover one edge's 64 channels (float4 each).
// L2-resident fp32 atomics (working set < 192MB L2).
__global__ void k_scatter(const long long* __restrict__ src, const long long* __restrict__ dst,
                          const float* __restrict__ xw, const float* __restrict__ dinv,
                          float* __restrict__ h, int E) {
    int gid = blockIdx.x * blockDim.x + threadIdx.x;
    if (gid >= E * 16) return;
    int e = gid >> 4;
    int c = (gid & 15) * 4;
    int s = (int)src[e];
    int d = (int)dst[e];
    float nm = dinv[s] * dinv[d];
    float4 m = *(const float4*)(xw + (size_t)s * D + c);
    float* hp = h + (size_t)d * D + c;
    atomicAdd(hp + 0, m.x * nm);
    atomicAdd(hp + 1, m.y * nm);
    atomicAdd(hp + 2, m.z * nm);
    atomicAdd(hp + 3, m.w * nm);
}

// Per-channel sum / sumsq: strided accumulation, LDS cross-group reduce, global atomics.
__global__ void k_bn_stats(const float* __restrict__ h, float* __restrict__ sums, int N) {
    const int c   = threadIdx.x & 63;
    const int grp = threadIdx.x >> 6;     // 0..3
    float s = 0.f, q = 0.f;
    for (int i = blockIdx.x * 4 + grp; i < N; i += gridDim.x * 4) {
        float v = h[(size_t)i * D + c];
        s += v; q += v * v;
    }
    __shared__ float bufS[256], bufQ[256];
    bufS[threadIdx.x] = s; bufQ[threadIdx.x] = q;
    __syncthreads();
    if (grp == 0) {
        s = bufS[c] + bufS[64 + c] + bufS[128 + c] + bufS[192 + c];
        q = bufQ[c] + bufQ[64 + c] + bufQ[128 + c] + bufQ[192 + c];
        atomicAdd(&sums[c], s);
        atomicAdd(&sums[64 + c], q);
    }
}

__global__ void k_bn_norm_relu(const float* __restrict__ h, const float* __restrict__ sums,
                               const float* __restrict__ gamma, const float* __restrict__ beta,
                               float* __restrict__ out, int N) {
    int gid = blockIdx.x * blockDim.x + threadIdx.x;
    if (gid >= N * 16) return;
    int i = gid >> 4;
    int c = (gid & 15) * 4;
    const float invN = 1.0f / (float)N;
    float4 hv = *(const float4*)(h + (size_t)i * D + c);
    float hvv[4] = {hv.x, hv.y, hv.z, hv.w};
    float o[4];
#pragma unroll
    for (int j = 0; j < 4; ++j) {
        float mu  = sums[c + j] * invN;
        float var = sums[64 + c + j] * invN - mu * mu;   // biased variance
        float r   = rsqrtf(var + BN_EPS);
        float v   = (hvv[j] - mu) * r * gamma[c + j] + beta[c + j];
        o[j] = v > 0.f ? v : 0.f;
    }
    *(float4*)(out + (size_t)i * D + c) = make_float4(o[0], o[1], o[2], o[3]);
}

extern "C" void kernel_launch(void* const* d_in, const int* in_sizes, int n_in,
                              void* d_out, int out_size, void* d_ws, size_t ws_size,
                              hipStream_t stream) {
    const float*     x     = (const float*)d_in[0];
    const long long* eidx  = (const long long*)d_in[1];   // int64 per reference
    const float*     W     = (const float*)d_in[2];
    const float*     bias  = (const float*)d_in[3];
    const float*     gamma = (const float*)d_in[4];
    const float*     beta  = (const float*)d_in[5];

    const int N = in_sizes[0] / D;        // 100000
    const int E = in_sizes[1] / 2;        // 800000
    const long long* src = eidx;
    const long long* dst = eidx + E;

    float* ws   = (float*)d_ws;
    float* dinv = ws;                                  // N (padded region 102400)
    float* xw   = ws + 102400;                         // N*64
    float* h    = xw + (size_t)N * D;                  // N*64
    float* sums = h  + (size_t)N * D;                  // 128

    const int T = 256;
    k_init      <<<(N + T - 1) / T, T, 0, stream>>>(dinv, sums, N);
    k_count_deg <<<(E + T - 1) / T, T, 0, stream>>>(dst, dinv, E);
    k_rsqrt     <<<(N + T - 1) / T, T, 0, stream>>>(dinv, N);
    k_gemm_wmma <<<(N + 127) / 128, T, 0, stream>>>(x, W, xw, N);
    k_init_h    <<<(N * 16 + T - 1) / T, T, 0, stream>>>(xw, dinv, bias, h, N);
    k_scatter   <<<(E * 16 + T - 1) / T, T, 0, stream>>>(src, dst, xw, dinv, h, E);
    k_bn_stats  <<<512, T, 0, stream>>>(h, sums, N);
    k_bn_norm_relu<<<(N * 16 + T - 1) / T, T, 0, stream>>>(h, sums, gamma, beta,
                                                           (float*)d_out, N);
}